// GCNConvModel_19146964206336
// MI455X (gfx1250) — compile-verified
//
#include <hip/hip_runtime.h>
#include <hip/hip_bf16.h>

typedef float v2f __attribute__((ext_vector_type(2)));
typedef float v4f __attribute__((ext_vector_type(4)));
typedef float v8f __attribute__((ext_vector_type(8)));

#define FDIM 64
#define BN_EPS 1e-5f

// ============================================================================
// WMMA GEMM: Y[M,64] = X[M,64] @ B(64,64) (+ bias), B derived from W:
//   TRANSB=false : B[k][n] = W[k*64+n]      (h = x @ conv_w)
//   TRANSB=true  : B[k][n] = W[n*64+k]      (out = pooled @ lin_w.T)
// Block = 128 threads = 4 waves; each wave computes a 16x64 output strip
// using V_WMMA_F32_16X16X4_F32 (16 K-steps x 4 N-tiles = 64 WMMAs / wave).
// LDS holds B K-pair interleaved: sB[(k>>1)*128 + n*2 + (k&1)] so every
// B fragment is one aligned ds_load_b64 straight into an even VGPR pair.
// ============================================================================
template <bool TRANSB, bool HASBIAS>
__global__ void __launch_bounds__(128)
gemm64_wmma(const float* __restrict__ X, const float* __restrict__ W,
            const float* __restrict__ bias, float* __restrict__ Y, int M)
{
    __shared__ float sB[FDIM * FDIM];
    for (int i = threadIdx.x; i < FDIM * FDIM; i += 128) {
        int r = i >> 6, c = i & 63;
        int k = TRANSB ? c : r;
        int n = TRANSB ? r : c;
        sB[((k >> 1) << 7) + (n << 1) + (k & 1)] = W[i];
    }
    __syncthreads();

    const int wave = threadIdx.x >> 5;
    const int lane = threadIdx.x & 31;
    const int m16  = lane & 15;
    const int grp  = lane >> 4;                 // 0 or 1 (K-halves of the frag)
    const int rowbase = blockIdx.x * 64 + wave * 16;

    // A-fragment source row for this lane; clamp (not mask) so EXEC stays all-1s
    int arow = rowbase + m16;
    if (arow > M - 1) arow = M - 1;
    const float* __restrict__ xrow = X + (size_t)arow * FDIM;

    v8f acc0 = {}, acc1 = {}, acc2 = {}, acc3 = {};
#pragma unroll
    for (int k0 = 0; k0 < FDIM; k0 += 4) {
        const int kd = k0 + 2 * grp;            // this lane's K pair: kd, kd+1 (even)
        v2f a;  a.x = xrow[kd];  a.y = xrow[kd + 1];
        const float* bp = sB + ((kd >> 1) << 7);
        v2f b0 = *(const v2f*)(bp + (m16 << 1));
        v2f b1 = *(const v2f*)(bp + ((16 + m16) << 1));
        v2f b2 = *(const v2f*)(bp + ((32 + m16) << 1));
        v2f b3 = *(const v2f*)(bp + ((48 + m16) << 1));
        acc0 = __builtin_amdgcn_wmma_f32_16x16x4_f32(false, a, false, b0, (short)0, acc0, false, false);
        acc1 = __builtin_amdgcn_wmma_f32_16x16x4_f32(false, a, false, b1, (short)0, acc1, false, false);
        acc2 = __builtin_amdgcn_wmma_f32_16x16x4_f32(false, a, false, b2, (short)0, acc2, false, false);
        acc3 = __builtin_amdgcn_wmma_f32_16x16x4_f32(false, a, false, b3, (short)0, acc3, false, false);
    }

    // C layout: VGPR r <-> row rowbase + r + 8*grp, col = ntile*16 + m16
#pragma unroll
    for (int r = 0; r < 8; ++r) {
        const int gr = rowbase + r + 8 * grp;
        if (gr < M) {
            float* yr = Y + (size_t)gr * FDIM;
            yr[m16]      = acc0[r] + (HASBIAS ? bias[m16]      : 0.f);
            yr[16 + m16] = acc1[r] + (HASBIAS ? bias[16 + m16] : 0.f);
            yr[32 + m16] = acc2[r] + (HASBIAS ? bias[32 + m16] : 0.f);
            yr[48 + m16] = acc3[r] + (HASBIAS ? bias[48 + m16] : 0.f);
        }
    }
}

// deg[dst] += 1 per edge (deg buffer pre-zeroed)
__global__ void deg_count(const long long* __restrict__ ei, float* __restrict__ deg, long long E)
{
    long long e = (long long)blockIdx.x * blockDim.x + threadIdx.x;
    if (e < E) atomicAdd(&deg[ei[E + e]], 1.0f);
}

// dinv = rsqrt(deg + 1)  (self loop), in place
__global__ void deg_to_dinv(float* __restrict__ d, int N)
{
    int i = blockIdx.x * blockDim.x + threadIdx.x;
    if (i < N) d[i] = rsqrtf(d[i] + 1.0f);
}

// agg[dst][f] += h[src][f] * dinv[src]*dinv[dst]
// One lane per (edge, feature-pair): b64 gather of h[src], two f32 atomics.
// A wave covers one edge's 64 features -> fully coalesced; dinv loads are
// wave-uniform (single address) so the coalescer merges them.
__global__ void edge_scatter(const float* __restrict__ h, const long long* __restrict__ ei,
                             const float* __restrict__ dinv, float* __restrict__ agg, long long E)
{
    long long idx = (long long)blockIdx.x * blockDim.x + threadIdx.x;
    if (idx >= E * (FDIM / 2)) return;
    long long e = idx >> 5;
    int f = (int)(idx & 31) << 1;
    long long s = ei[e], d = ei[E + e];
    float w = dinv[s] * dinv[d];
    v2f hv = *(const v2f*)(h + s * FDIM + f);
    float* ap = agg + d * FDIM + f;
    atomicAdd(ap,     hv.x * w);
    atomicAdd(ap + 1, hv.y * w);
}

// agg += h*dinv^2 + conv_b (in place, float4) and accumulate per-feature
// sum/sumsq. Grid stride (in float4 units) is a multiple of 16 so each thread
// owns a fixed 4-feature column group.
__global__ void self_bias_stats(float* __restrict__ agg, const float* __restrict__ h,
                                const float* __restrict__ dinv, const float* __restrict__ cb,
                                float* __restrict__ bsum, float* __restrict__ bsq, int N)
{
    const long long total4 = (long long)N * (FDIM / 4);
    const long long step   = (long long)gridDim.x * blockDim.x;   // 1024*256, %16==0
    long long idx = (long long)blockIdx.x * blockDim.x + threadIdx.x;
    const int f0 = (int)((idx << 2) & 63);
    const v4f bf = *(const v4f*)(cb + f0);
    v4f s = {}, s2 = {};
    for (; idx < total4; idx += step) {
        int i = (int)(idx >> 4);
        float di = dinv[i];
        float sn = di * di;
        v4f av = *(const v4f*)(agg + (idx << 2));
        v4f hv = *(const v4f*)(h + (idx << 2));
        v4f v;
#pragma unroll
        for (int j = 0; j < 4; ++j) v[j] = av[j] + hv[j] * sn + bf[j];
        *(v4f*)(agg + (idx << 2)) = v;
#pragma unroll
        for (int j = 0; j < 4; ++j) { s[j] += v[j]; s2[j] += v[j] * v[j]; }
    }
#pragma unroll
    for (int j = 0; j < 4; ++j) {
        atomicAdd(&bsum[f0 + j], s[j]);
        atomicAdd(&bsq[f0 + j],  s2[j]);
    }
}

__global__ void bn_finalize(const float* __restrict__ bsum, const float* __restrict__ bsq,
                            float* __restrict__ mean, float* __restrict__ inv, int N)
{
    int f = threadIdx.x;
    float m = bsum[f] / (float)N;
    float v = bsq[f] / (float)N - m * m;       // biased variance
    mean[f] = m;
    inv[f]  = rsqrtf(v + BN_EPS);
}

// x = relu(gamma * (x - mean)*inv + beta), in place, float4
__global__ void bn_apply(float* __restrict__ x, const float* __restrict__ mean,
                         const float* __restrict__ inv, const float* __restrict__ g,
                         const float* __restrict__ b, long long total4)
{
    long long idx = (long long)blockIdx.x * blockDim.x + threadIdx.x;
    if (idx >= total4) return;
    const int f0 = (int)((idx << 2) & 63);
    v4f mv = *(const v4f*)(mean + f0);
    v4f iv = *(const v4f*)(inv + f0);
    v4f gv = *(const v4f*)(g + f0);
    v4f bv = *(const v4f*)(b + f0);
    v4f v = *(v4f*)(x + (idx << 2));
#pragma unroll
    for (int j = 0; j < 4; ++j) {
        float t = (v[j] - mv[j]) * iv[j];
        t = gv[j] * t + bv[j];
        v[j] = t > 0.f ? t : 0.f;
    }
    *(v4f*)(x + (idx << 2)) = v;
}

__global__ void pool_acc(const float* __restrict__ x, const long long* __restrict__ batch,
                         float* __restrict__ pooled, float* __restrict__ counts, int N)
{
    long long idx = (long long)blockIdx.x * blockDim.x + threadIdx.x;
    if (idx >= (long long)N * FDIM) return;
    int i = (int)(idx >> 6);
    int f = (int)(idx & 63);
    long long g = batch[i];
    atomicAdd(&pooled[g * FDIM + f], x[idx]);
    if (f == 0) atomicAdd(&counts[g], 1.0f);
}

__global__ void pool_div(float* __restrict__ pooled, const float* __restrict__ counts, int GF)
{
    int idx = blockIdx.x * blockDim.x + threadIdx.x;
    if (idx >= GF) return;
    float c = counts[idx >> 6];
    pooled[idx] /= (c < 1.0f ? 1.0f : c);
}

extern "C" void kernel_launch(void* const* d_in, const int* in_sizes, int n_in,
                              void* d_out, int out_size, void* d_ws, size_t ws_size,
                              hipStream_t stream)
{
    (void)n_in; (void)ws_size;
    const float*     x      = (const float*)d_in[0];
    const long long* ei     = (const long long*)d_in[1];
    const long long* batch  = (const long long*)d_in[2];
    const float*     conv_w = (const float*)d_in[3];
    const float*     conv_b = (const float*)d_in[4];
    const float*     bn_g   = (const float*)d_in[5];
    const float*     bn_b   = (const float*)d_in[6];
    const float*     lin_w  = (const float*)d_in[7];
    const float*     lin_b  = (const float*)d_in[8];
    float*           out    = (float*)d_out;

    const int       N  = in_sizes[0] / FDIM;
    const long long E  = in_sizes[1] / 2;
    const int       L  = in_sizes[3] / (FDIM * FDIM);
    const int       G  = out_size / FDIM;
    const long long NF = (long long)N * FDIM;

    // workspace layout (floats)
    float* A      = (float*)d_ws;            // h = x @ W         [N,64]
    float* B      = A + NF;                  // agg / activations [N,64]
    float* dinv   = B + NF;                  // deg -> deg^{-1/2} [N]
    float* bsum   = dinv + N;                // [64]
    float* bsq    = bsum + FDIM;             // [64]
    float* mean   = bsq + FDIM;              // [64]
    float* inv    = mean + FDIM;             // [64]
    float* pooled = inv + FDIM;              // [G,64]
    float* counts = pooled + (long long)G * FDIM;  // [G]

    // degree -> dinv
    hipMemsetAsync(dinv, 0, (size_t)N * sizeof(float), stream);
    deg_count<<<(int)((E + 255) / 256), 256, 0, stream>>>(ei, dinv, E);
    deg_to_dinv<<<(N + 255) / 256, 256, 0, stream>>>(dinv, N);

    const float* cur = x;
    for (int l = 0; l < L; ++l) {
        gemm64_wmma<false, false><<<(N + 63) / 64, 128, 0, stream>>>(
            cur, conv_w + (size_t)l * FDIM * FDIM, nullptr, A, N);
        hipMemsetAsync(B, 0, (size_t)NF * sizeof(float), stream);
        hipMemsetAsync(bsum, 0, 2 * FDIM * sizeof(float), stream);   // bsum+bsq
        edge_scatter<<<(int)((E * (FDIM / 2) + 255) / 256), 256, 0, stream>>>(A, ei, dinv, B, E);
        self_bias_stats<<<1024, 256, 0, stream>>>(B, A, dinv, conv_b + (size_t)l * FDIM,
                                                  bsum, bsq, N);
        bn_finalize<<<1, FDIM, 0, stream>>>(bsum, bsq, mean, inv, N);
        bn_apply<<<(int)((NF / 4 + 255) / 256), 256, 0, stream>>>(
            B, mean, inv, bn_g + (size_t)l * FDIM, bn_b + (size_t)l * FDIM, NF / 4);
        cur = B;
    }

    // global mean pool + final linear (WMMA, B = lin_w^T, bias = lin_b)
    hipMemsetAsync(pooled, 0, (size_t)(G * FDIM + G) * sizeof(float), stream);
    pool_acc<<<(int)((NF + 255) / 256), 256, 0, stream>>>(B, batch, pooled, counts, N);
    pool_div<<<(G * FDIM + 255) / 256, 256, 0, stream>>>(pooled, counts, G * FDIM);
    gemm64_wmma<true, true><<<(G + 63) / 64, 128, 0, stream>>>(pooled, lin_w, lin_b, out, G);
}